// WindowAttention_87797721465072
// MI455X (gfx1250) — compile-verified
//
#include <hip/hip_runtime.h>

typedef __attribute__((ext_vector_type(2))) float v2f;
typedef __attribute__((ext_vector_type(8))) float v8f;

namespace {
constexpr int   kN     = 49;    // tokens per window
constexpr int   kDim   = 256;
constexpr int   kHeads = 8;
constexpr int   kHd    = 32;
constexpr int   kQCols = 768;
constexpr float kScale = 0.17677669529663687f;  // 32^-0.5

// LDS layout (units: floats). Strides chosen for b64 alignment + bank spread.
constexpr int XS_S = 260;                       // x / ctx tile stride, 64 rows
constexpr int QS_S = 771;                       // qkv tile stride, 52 rows (49..51 zeroed)
constexpr int PS_S = 68;                        // per-wave P strip stride, 16 rows

constexpr int XS_OFF   = 0;
constexpr int QKV_OFF  = XS_OFF + 64 * XS_S;    // 16640
constexpr int P_OFF    = QKV_OFF + 52 * QS_S;   // 56732 (even -> 8B aligned)
constexpr int BIAS_OFF = P_OFF + kHeads * 16 * PS_S;  // 65436
constexpr int MASK_OFF = BIAS_OFF + 169 * kHeads;     // 66788
constexpr int SMEM_FLOATS = MASK_OFF + kN * kN + 3;   // 69192
constexpr int SMEM_BYTES  = SMEM_FLOATS * 4;          // 276768 B  (< 320 KB/WGP)
}  // namespace

__device__ __forceinline__ v8f wmma4(v2f a, v2f b, v8f c) {
  // D = A(16x4) * B(4x16) + C, fp32 exact
  return __builtin_amdgcn_wmma_f32_16x16x4_f32(
      /*neg_a=*/false, a, /*neg_b=*/false, b,
      /*c_mod=*/(short)0, c, /*reuse_a=*/false, /*reuse_b=*/false);
}

// Reductions across the 16-lane half that holds one softmax row
// (xor masks < 16 never cross the half boundary on wave32).
__device__ __forceinline__ float hmax16(float v) {
  v = fmaxf(v, __shfl_xor(v, 1, 32));
  v = fmaxf(v, __shfl_xor(v, 2, 32));
  v = fmaxf(v, __shfl_xor(v, 4, 32));
  v = fmaxf(v, __shfl_xor(v, 8, 32));
  return v;
}
__device__ __forceinline__ float hadd16(float v) {
  v += __shfl_xor(v, 1, 32);
  v += __shfl_xor(v, 2, 32);
  v += __shfl_xor(v, 4, 32);
  v += __shfl_xor(v, 8, 32);
  return v;
}

__global__ __launch_bounds__(256) void winattn_fused(
    const float* __restrict__ x, const int* __restrict__ mask,
    const float* __restrict__ qkv_w, const float* __restrict__ qkv_b,
    const float* __restrict__ proj_w, const float* __restrict__ proj_b,
    const float* __restrict__ relb, float* __restrict__ out) {
  extern __shared__ float lds[];
  float* xs = lds + XS_OFF;    // x tile, later reused as ctx tile
  float* qs = lds + QKV_OFF;   // qkv[52][QS_S], col = h*96 + d*3 + {q,k,v}
  float* ps = lds + P_OFF;     // per-wave prob strip [16][PS_S]
  float* bs = lds + BIAS_OFF;  // rel-pos bias table [169][8]
  float* ms = lds + MASK_OFF;  // precomputed (1-mask)*-1e10 [49][49]

  const int tid  = threadIdx.x;
  const int lane = tid & 31;
  const int hl   = lane >> 4;   // which 16-lane half
  const int l16  = lane & 15;
  const int wave = tid >> 5;    // 8 waves
  const int blk  = blockIdx.x;  // one window per block

  // ---------------- Phase 0: stage window data into LDS ----------------
  {
    const float4* xv = (const float4*)(x + (size_t)blk * (kN * kDim));
    for (int i = tid; i < (kN * kDim) / 4; i += 256) {
      float4 v  = xv[i];
      int   row = i >> 6;            // 64 float4 per 256-wide row
      int   col = (i & 63) << 2;
      *(float4*)&xs[row * XS_S + col] = v;
    }
    for (int i = tid; i < 3 * QS_S; i += 256)  // zero K-pad rows 49..51
      qs[49 * QS_S + i] = 0.0f;
    for (int i = tid; i < 169 * kHeads; i += 256)
      bs[i] = relb[i];
    const int* mw = mask + (size_t)blk * (kN * kN);
    for (int i = tid; i < kN * kN; i += 256)
      ms[i] = (1.0f - (float)mw[i]) * -1.0e10f;
  }
  __syncthreads();

  // ---------------- Phase 1: QKV = xs @ qkv_w + qkv_b ----------------
  // Wave w produces columns [96w, 96w+96) == q/k/v of head w.
  {
    for (int ct = 0; ct < 6; ++ct) {
      const int n = wave * 96 + ct * 16 + l16;  // B/C column
      v8f acc0 = {}, acc1 = {}, acc2 = {}, acc3 = {};
      for (int k0 = 0; k0 < kDim; k0 += 4) {
        const int kk = k0 + 2 * hl;
        v2f bf;
        bf.x = qkv_w[(size_t)kk * kQCols + n];
        bf.y = qkv_w[(size_t)(kk + 1) * kQCols + n];
        v2f a0 = *(const v2f*)&xs[(0 * 16 + l16) * XS_S + kk];
        v2f a1 = *(const v2f*)&xs[(1 * 16 + l16) * XS_S + kk];
        v2f a2 = *(const v2f*)&xs[(2 * 16 + l16) * XS_S + kk];
        v2f a3 = *(const v2f*)&xs[(3 * 16 + l16) * XS_S + kk];
        acc0 = wmma4(a0, bf, acc0);
        acc1 = wmma4(a1, bf, acc1);
        acc2 = wmma4(a2, bf, acc2);
        acc3 = wmma4(a3, bf, acc3);
      }
      const float bb = qkv_b[n];
      v8f accs[4] = {acc0, acc1, acc2, acc3};
#pragma unroll
      for (int rt = 0; rt < 4; ++rt)
#pragma unroll
        for (int r = 0; r < 8; ++r) {
          const int tok = rt * 16 + r + 8 * hl;
          if (tok < kN) qs[tok * QS_S + n] = accs[rt][r] + bb;
        }
    }
  }
  __syncthreads();  // protects xs -> ctx reuse below

  // ---------------- Phase 2: attention, head h == wave ----------------
  {
    const int h  = wave;
    const int qb = h * 96;
    float*    pw = ps + wave * 16 * PS_S;
    for (int rt = 0; rt < 4; ++rt) {
      // S = Q @ K^T : 16-row strip x 64 padded cols
      v8f s0 = {}, s1 = {}, s2 = {}, s3 = {};
      for (int k0 = 0; k0 < kHd; k0 += 4) {
        const int d = k0 + 2 * hl;
        v2f a;
        a.x = qs[(rt * 16 + l16) * QS_S + qb + d * 3];
        a.y = qs[(rt * 16 + l16) * QS_S + qb + (d + 1) * 3];
        v2f b0, b1, b2, b3;
        b0.x = qs[(0 * 16 + l16) * QS_S + qb + d * 3 + 1];
        b0.y = qs[(0 * 16 + l16) * QS_S + qb + (d + 1) * 3 + 1];
        b1.x = qs[(1 * 16 + l16) * QS_S + qb + d * 3 + 1];
        b1.y = qs[(1 * 16 + l16) * QS_S + qb + (d + 1) * 3 + 1];
        b2.x = qs[(2 * 16 + l16) * QS_S + qb + d * 3 + 1];
        b2.y = qs[(2 * 16 + l16) * QS_S + qb + (d + 1) * 3 + 1];
        b3.x = qs[(3 * 16 + l16) * QS_S + qb + d * 3 + 1];
        b3.y = qs[(3 * 16 + l16) * QS_S + qb + (d + 1) * 3 + 1];
        s0 = wmma4(a, b0, s0);
        s1 = wmma4(a, b1, s1);
        s2 = wmma4(a, b2, s2);
        s3 = wmma4(a, b3, s3);
      }
      v8f sarr[4] = {s0, s1, s2, s3};
      // scale + rel-pos bias + mask, then row softmax; write P strip to LDS
#pragma unroll
      for (int r = 0; r < 8; ++r) {
        const int qt = rt * 16 + r + 8 * hl;
        float vals[4];
        float vmax = -3.0e38f;
#pragma unroll
        for (int ct = 0; ct < 4; ++ct) {
          const int kt = ct * 16 + l16;
          float     v;
          if (qt < kN && kt < kN) {
            const int di = qt / 7 - kt / 7 + 6;
            const int dj = qt % 7 - kt % 7 + 6;
            v = sarr[ct][r] * kScale + bs[(di * 13 + dj) * kHeads + h] +
                ms[qt * kN + kt];
          } else {
            v = -1.0e30f;  // padded key cols -> zero probability
          }
          vals[ct] = v;
          vmax     = fmaxf(vmax, v);
        }
        vmax      = hmax16(vmax);
        float sum = 0.0f;
#pragma unroll
        for (int ct = 0; ct < 4; ++ct) {
          const float p = __expf(vals[ct] - vmax);
          vals[ct]      = p;
          sum += p;
        }
        sum             = hadd16(sum);
        const float inv = 1.0f / sum;
#pragma unroll
        for (int ct = 0; ct < 4; ++ct)
          pw[(r + 8 * hl) * PS_S + ct * 16 + l16] = vals[ct] * inv;
      }
      // O = P @ V  (K padded to 52; qs rows 49..51 are zero)
#pragma unroll
      for (int cto = 0; cto < 2; ++cto) {
        v8f o = {};
        for (int k0 = 0; k0 < 52; k0 += 4) {
          const int kk = k0 + 2 * hl;
          v2f a = *(const v2f*)&pw[l16 * PS_S + kk];
          v2f b;
          b.x = qs[kk * QS_S + qb + (cto * 16 + l16) * 3 + 2];
          b.y = qs[(kk + 1) * QS_S + qb + (cto * 16 + l16) * 3 + 2];
          o = wmma4(a, b, o);
        }
#pragma unroll
        for (int r = 0; r < 8; ++r) {
          const int tok = rt * 16 + r + 8 * hl;
          if (tok < kN)
            xs[tok * XS_S + h * kHd + cto * 16 + l16] = o[r];  // ctx tile
        }
      }
    }
  }
  __syncthreads();

  // ---------------- Phase 3: out = ctx @ proj_w + proj_b ----------------
  {
    float* ob = out + (size_t)blk * (kN * kDim);
    for (int cto = 0; cto < 2; ++cto) {
      const int n = wave * 32 + cto * 16 + l16;
      v8f acc0 = {}, acc1 = {}, acc2 = {}, acc3 = {};
      for (int k0 = 0; k0 < kDim; k0 += 4) {
        const int kk = k0 + 2 * hl;
        v2f bf;
        bf.x = proj_w[(size_t)kk * kDim + n];
        bf.y = proj_w[(size_t)(kk + 1) * kDim + n];
        v2f a0 = *(const v2f*)&xs[(0 * 16 + l16) * XS_S + kk];
        v2f a1 = *(const v2f*)&xs[(1 * 16 + l16) * XS_S + kk];
        v2f a2 = *(const v2f*)&xs[(2 * 16 + l16) * XS_S + kk];
        v2f a3 = *(const v2f*)&xs[(3 * 16 + l16) * XS_S + kk];
        acc0 = wmma4(a0, bf, acc0);
        acc1 = wmma4(a1, bf, acc1);
        acc2 = wmma4(a2, bf, acc2);
        acc3 = wmma4(a3, bf, acc3);
      }
      const float pb      = proj_b[n];
      v8f         accs[4] = {acc0, acc1, acc2, acc3};
#pragma unroll
      for (int rt = 0; rt < 4; ++rt)
#pragma unroll
        for (int r = 0; r < 8; ++r) {
          const int tok = rt * 16 + r + 8 * hl;
          if (tok < kN) ob[tok * kDim + n] = accs[rt][r] + pb;
        }
    }
  }
}

extern "C" void kernel_launch(void* const* d_in, const int* in_sizes, int n_in,
                              void* d_out, int out_size, void* d_ws,
                              size_t ws_size, hipStream_t stream) {
  (void)in_sizes; (void)n_in; (void)out_size; (void)d_ws; (void)ws_size;
  const float* x      = (const float*)d_in[0];
  const int*   mask   = (const int*)d_in[1];
  const float* qkv_w  = (const float*)d_in[2];
  const float* qkv_b  = (const float*)d_in[3];
  const float* proj_w = (const float*)d_in[4];
  const float* proj_b = (const float*)d_in[5];
  const float* relb   = (const float*)d_in[6];
  float*       out    = (float*)d_out;

  (void)hipFuncSetAttribute((const void*)winattn_fused,
                            hipFuncAttributeMaxDynamicSharedMemorySize,
                            SMEM_BYTES);
  winattn_fused<<<dim3(4096), dim3(256), SMEM_BYTES, stream>>>(
      x, mask, qkv_w, qkv_b, proj_w, proj_b, relb, out);
}